// RoIPooling_20057497272712
// MI455X (gfx1250) — compile-verified
//
#include <hip/hip_runtime.h>

// RoI crop-and-resize (bilinear), 8x64x64x256 f32 feature map, 2400 ROIs -> 7x7x256 tiles.
// Memory-bound gather kernel: ~33.5 MB fmap read (L2-resident) + ~120 MB streamed output
// => ~6.6us at 23.3 TB/s HBM. WMMA intentionally NOT used: the 4-corner operand varies
// per output position, so the op does not factor into a wave-shared B matrix; densifying
// (out = Wy*F*Wx^T per ROI) would multiply L2 traffic ~4x and change numerics.
// Optimal gfx1250 path: coalesced global_load_b128 gathers + non-temporal global_store_b128,
// with all per-ROI state (box, batch index) kept scalar via a 2-D grid (blockIdx.y = roi).

typedef float f4 __attribute__((ext_vector_type(4)));

namespace {
constexpr int kB  = 8;
constexpr int kH  = 64;
constexpr int kW  = 64;
constexpr int kC  = 256;     // channels = 64 float4 groups
constexpr int kN  = 300;     // ROIs per batch image
constexpr int kPH = 7;
constexpr int kPW = 7;
constexpr int kPos = kPH * kPW;                   // 49 positions per ROI
constexpr int kPosPerBlock = 4;                   // 256 threads = 4 groups of 64 lanes
constexpr int kBlocksX = (kPos + kPosPerBlock - 1) / kPosPerBlock;  // 13
}

__global__ __launch_bounds__(256) void roi_crop_resize_kernel(
    const float* __restrict__ fmap,    // [B,H,W,C]
    const float* __restrict__ boxes,   // [B*N, 4] = (y_lo, x_lo, y_hi, x_hi)
    float* __restrict__ out)           // [B*N, PH, PW, C]
{
    const int roi   = blockIdx.y;                       // uniform per block -> SGPR
    const int group = threadIdx.x >> 6;                 // 0..3  (64-lane group)
    const int lane  = threadIdx.x & 63;                 // 0..63 -> channels lane*4 .. lane*4+3
    const int p     = blockIdx.x * kPosPerBlock + group;  // position within ROI
    if (p >= kPos) return;

    const int py = p / kPW;            // p in [0,49): single cheap magic-mul
    const int px = p - py * kPW;
    const int b  = roi / kN;           // uniform -> scalar division, once per block

    // Box is uniform across the whole block: one s_load_b128.
    const f4 box = *(const f4*)(boxes + (size_t)roi * 4);
    const float lo_y = box.x, lo_x = box.y, hi_y = box.z, hi_x = box.w;

    // Reference math: t = i/(P-1); coord = (lo + t*(hi-lo)) * (size-1)
    const float ty = (float)py * (1.0f / 6.0f);
    const float tx = (float)px * (1.0f / 6.0f);
    const float yy = (lo_y + ty * (hi_y - lo_y)) * (float)(kH - 1);
    const float xx = (lo_x + tx * (hi_x - lo_x)) * (float)(kW - 1);

    const bool valid = (yy >= 0.0f) & (yy <= (float)(kH - 1)) &
                       (xx >= 0.0f) & (xx <= (float)(kW - 1));

    const float y0f = floorf(yy);
    const float x0f = floorf(xx);
    const float wy  = yy - y0f;
    const float wx  = xx - x0f;

    const int y0 = (int)fminf(fmaxf(y0f,        0.0f), (float)(kH - 1));
    const int y1 = (int)fminf(fmaxf(y0f + 1.0f, 0.0f), (float)(kH - 1));
    const int x0 = (int)fminf(fmaxf(x0f,        0.0f), (float)(kW - 1));
    const int x1 = (int)fminf(fmaxf(x0f + 1.0f, 0.0f), (float)(kW - 1));

    // Corner row bases: each row segment is 256 f32 = 1KB contiguous.
    const size_t ib = (size_t)b * kH * kW * kC;
    const f4* r00 = (const f4*)(fmap + ib + ((size_t)y0 * kW + x0) * kC);
    const f4* r01 = (const f4*)(fmap + ib + ((size_t)y0 * kW + x1) * kC);
    const f4* r10 = (const f4*)(fmap + ib + ((size_t)y1 * kW + x0) * kC);
    const f4* r11 = (const f4*)(fmap + ib + ((size_t)y1 * kW + x1) * kC);

    // 4x coalesced global_load_b128 gathers (64 lanes x 16B = 1KB each).
    const f4 tl = r00[lane];
    const f4 tr = r01[lane];
    const f4 bl = r10[lane];
    const f4 br = r11[lane];

    // Bilinear blend, same association order as the reference.
    f4 top = tl + (tr - tl) * wx;
    f4 bot = bl + (br - bl) * wx;
    f4 val = top + (bot - top) * wy;
    if (!valid) {
        val = f4{0.0f, 0.0f, 0.0f, 0.0f};
    }

    // Write-once 120MB output: non-temporal b128 store keeps L2 for the feature map.
    f4* o = (f4*)(out + ((size_t)roi * kPos + p) * kC);
    __builtin_nontemporal_store(val, o + lane);
}

extern "C" void kernel_launch(void* const* d_in, const int* in_sizes, int n_in,
                              void* d_out, int out_size, void* d_ws, size_t ws_size,
                              hipStream_t stream) {
    const float* fmap  = (const float*)d_in[0];   // (8,64,64,256) f32
    const float* boxes = (const float*)d_in[1];   // (8,300,4)     f32
    float* out = (float*)d_out;                   // (8,300,7,7,256) f32

    dim3 grid(kBlocksX, kB * kN);                 // (13, 2400)
    roi_crop_resize_kernel<<<grid, 256, 0, stream>>>(fmap, boxes, out);
}